// AttentionAggregator_35742717837625
// MI455X (gfx1250) — compile-verified
//
#include <hip/hip_runtime.h>
#include <hip/hip_bf16.h>

// Problem constants (fixed by the reference)
constexpr int kN   = 100000;   // nodes
constexpr int kE   = 1600000;  // edges
constexpr int kDIN = 256;
constexpr int kH   = 4;
constexpr int kDH  = 32;
constexpr int kC   = 256;      // total output columns = (ORDER+1)*H*DH
constexpr float kNegSlope = 0.2f;
constexpr float kEps = 1e-9f;

typedef __attribute__((ext_vector_type(2))) float v2f;
typedef __attribute__((ext_vector_type(8))) float v8f;

// ---------------------------------------------------------------------------
// Kernel 1: feat[o][n][h*32+k] = relu(feat_in @ Wcat + b)
// Wcat column c = o*128 + h*32 + k maps to W[o][h][d][k] (stride 32 over d).
// One wave32 computes one 16x16 tile via V_WMMA_F32_16X16X4_F32 (K=256 -> 64 steps).
// ---------------------------------------------------------------------------
__global__ __launch_bounds__(256) void gemm_relu_kernel(
    const float* __restrict__ feat_in, const float* __restrict__ W,
    const float* __restrict__ bias, float* __restrict__ feat_ws)
{
  const int lane   = threadIdx.x & 31;
  const int wave   = threadIdx.x >> 5;
  const int gid    = blockIdx.x * 8 + wave;   // 0..99999 tiles
  const int m_tile = gid >> 4;                // 6250 row tiles
  const int n_tile = gid & 15;                // 16 col tiles
  const int row0   = m_tile * 16;
  const int col0   = n_tile * 16;

  const int lane_lo = lane & 15;
  const int khalf   = (lane >> 4) * 2;        // upper half-wave covers K+2,K+3

  // B-fragment addressing: column c of Wcat, element d is W[(c>>5)*8192 + d*32 + (c&31)]
  const int  c    = col0 + lane_lo;
  const long wcol = (long)(c >> 5) * (kDIN * kDH) + (c & 31);
  const float* arow = feat_in + (long)(row0 + lane_lo) * kDIN + khalf;

  v8f acc = {};
#pragma unroll 8
  for (int kk = 0; kk < kDIN; kk += 4) {
    v2f a = *(const v2f*)(arow + kk);                 // A: 16x4 fp32 fragment
    v2f bf;
    bf.x = W[wcol + (long)(kk + khalf)     * kDH];    // B: 4x16 fp32 fragment
    bf.y = W[wcol + (long)(kk + khalf + 1) * kDH];
    acc = __builtin_amdgcn_wmma_f32_16x16x4_f32(
        /*neg_a=*/false, a, /*neg_b=*/false, bf,
        /*c_mod=*/(short)0, acc, /*reuse_a=*/false, /*reuse_b=*/false);
  }

  // Epilogue: bias (b_flat[c] == b[o][h][k]) + relu, store to feat_ws[o][n][c&127]
  const float bv    = bias[c];
  const int   o     = c >> 7;
  const int   cw    = c & 127;
  const int   rbase = row0 + ((lane >> 4) << 3);      // +8 rows for upper half
#pragma unroll
  for (int j = 0; j < 8; ++j) {
    float v = fmaxf(acc[j] + bv, 0.0f);
    feat_ws[((long)o * kN + (rbase + j)) * 128 + cw] = v;
  }
}

// ---------------------------------------------------------------------------
// Kernel 2: per-(node, head) attention logits with leaky-relu.
//   att_s[n*4+h] = leaky(feat_self[h,n,:] . att[0,h,0:32])
//   att_n[n*4+h] = leaky(feat_nei [h,n,:] . att[0,h,32:64])
// ---------------------------------------------------------------------------
__global__ __launch_bounds__(256) void att_kernel(
    const float* __restrict__ feat_ws, const float* __restrict__ att,
    float* __restrict__ att_s, float* __restrict__ att_n)
{
  int t = blockIdx.x * blockDim.x + threadIdx.x;
  if (t >= kN * kH) return;
  int h = t & 3;
  int n = t >> 2;
  const float* f0 = feat_ws + (long)n * 128 + h * kDH;
  const float* f1 = feat_ws + ((long)kN + n) * 128 + h * kDH;
  const float* as = att + h * (2 * kDH);
  const float* an = as + kDH;
  float d0 = 0.f, d1 = 0.f;
#pragma unroll
  for (int k = 0; k < kDH; ++k) {
    d0 += f0[k] * as[k];
    d1 += f1[k] * an[k];
  }
  d0 = (d0 >= 0.f) ? d0 : kNegSlope * d0;
  d1 = (d1 >= 0.f) ? d1 : kNegSlope * d1;
  att_s[t] = d0;
  att_n[t] = d1;
}

// ---------------------------------------------------------------------------
// Kernel 3: edge scatter. 128 threads per edge (thread k covers column k).
//   agg[row[e]][k] += (att_s[row,h] + att_n[col,h]) * adj[e] * feat_nei[col][k]
// unsafeAtomicAdd -> global_atomic_add_f32 (hardware fp32 atomic in L2).
// ---------------------------------------------------------------------------
__global__ __launch_bounds__(256) void edge_kernel(
    const int* __restrict__ row, const int* __restrict__ col,
    const float* __restrict__ adj, const float* __restrict__ att_s,
    const float* __restrict__ att_n, const float* __restrict__ feat_ws,
    float* __restrict__ agg)
{
  long t = (long)blockIdx.x * 256 + threadIdx.x;
  int  e = (int)(t >> 7);
  int  k = (int)(t & 127);
  if (e >= kE) return;
  int r  = row[e];
  int cc = col[e];
  int h  = k >> 5;
  float w = (att_s[r * 4 + h] + att_n[cc * 4 + h]) * adj[e];
  float v = w * feat_ws[((long)kN + cc) * 128 + k];
  unsafeAtomicAdd(&agg[(long)r * 128 + k], v);
}

// ---------------------------------------------------------------------------
// Kernel 4: per-(order, head, node) layernorm over DH=32 (one wave32 each),
// then write out[n][o*128 + h*32 + k].
// ---------------------------------------------------------------------------
__global__ __launch_bounds__(256) void ln_kernel(
    const float* __restrict__ feat_ws, const float* __restrict__ agg,
    const float* __restrict__ scale, const float* __restrict__ offset,
    float* __restrict__ out)
{
  int wg   = blockIdx.x * 8 + (threadIdx.x >> 5);
  int lane = threadIdx.x & 31;
  int n    = wg >> 3;
  int rem  = wg & 7;
  int o    = rem >> 2;
  int h    = rem & 3;
  const float* src = (o == 0) ? (feat_ws + (long)n * 128) : (agg + (long)n * 128);
  float x = src[h * kDH + lane];
  float s1 = x, s2 = x * x;
#pragma unroll
  for (int off = 16; off > 0; off >>= 1) {
    s1 += __shfl_xor(s1, off, 32);
    s2 += __shfl_xor(s2, off, 32);
  }
  float mean = s1 * (1.0f / kDH);
  float var  = s2 * (1.0f / kDH) - mean * mean;
  float inv  = rsqrtf(var + kEps);
  int cidx = o * 128 + h * kDH + lane;
  out[(long)n * kC + cidx] = (x - mean) * scale[cidx] * inv + offset[cidx];
}

// ---------------------------------------------------------------------------
extern "C" void kernel_launch(void* const* d_in, const int* in_sizes, int n_in,
                              void* d_out, int out_size, void* d_ws, size_t ws_size,
                              hipStream_t stream) {
  const float* feat_in = (const float*)d_in[0];
  const int*   row     = (const int*)d_in[1];
  const int*   col     = (const int*)d_in[2];
  const float* adj     = (const float*)d_in[3];
  const float* W       = (const float*)d_in[4];
  const float* b       = (const float*)d_in[5];
  const float* att     = (const float*)d_in[6];
  const float* scale   = (const float*)d_in[7];
  const float* offset  = (const float*)d_in[8];
  float*       out     = (float*)d_out;

  // Workspace layout (floats)
  float* ws      = (float*)d_ws;
  float* feat_ws = ws;                                   // 2 * N * 128
  float* att_s   = feat_ws + (size_t)2 * kN * 128;       // N * 4
  float* att_n   = att_s + (size_t)kN * kH;              // N * 4
  float* agg     = att_n + (size_t)kN * kH;              // N * 128

  hipMemsetAsync(agg, 0, (size_t)kN * 128 * sizeof(float), stream);

  // 100000 16x16 tiles, 8 waves (8 tiles) per 256-thread block
  gemm_relu_kernel<<<12500, 256, 0, stream>>>(feat_in, W, b, feat_ws);

  att_kernel<<<(kN * kH + 255) / 256, 256, 0, stream>>>(feat_ws, att, att_s, att_n);

  // 2 edges per 256-thread block
  edge_kernel<<<(kE * 128 + 255) / 256, 256, 0, stream>>>(row, col, adj, att_s,
                                                          att_n, feat_ws, agg);

  // 800000 wave-sized LN jobs, 8 per block
  ln_kernel<<<kN, 256, 0, stream>>>(feat_ws, agg, scale, offset, out);
}